// Lag_PP_zero_28570122453265
// MI455X (gfx1250) — compile-verified
//
#include <hip/hip_runtime.h>

#define LDIM  1024
#define BATCH 8
#define TS    32
#define NTILE 32          // LDIM / TS
#define NPAIR 528         // NTILE*(NTILE+1)/2
#define NSTEP 10

typedef float v2f __attribute__((ext_vector_type(2)));
typedef float v8f __attribute__((ext_vector_type(8)));

__device__ __forceinline__ float sigf(float z) {
    return 1.0f / (1.0f + __expf(-z));
}

// One block = one (batch, tile-pair {ti<=tj}). Handles tiles (I,J) and (J,I).
// t == -1 : init phase (compute ah0 from u, write state to out slice 0, row sums)
// t >= 0  : step t (read state from slice t, write a_new to slice t (NT stores),
//           write next state to slice t+1 (t<9), emit row-sum partials to P)
__global__ __launch_bounds__(256)
void fused_step_kernel(const float* __restrict__ u,
                       const float4* __restrict__ x4,
                       float* __restrict__ out,
                       const float* __restrict__ rvec,
                       float* __restrict__ P,
                       int t, float alpha_t, float alpha_next)
{
    __shared__ float ldsU2[TS][TS + 1];   // u2 of tile (J,I), row-major in tile2 coords
    __shared__ float ldsA2[TS][TS + 1];   // ah state of tile (J,I)
    __shared__ float ldsV [TS][TS + 1];   // a_new values of tile (I,J)
    __shared__ float ldsN [TS][TS + 1];   // new ah of tile (J,I) (staged for coalesced store)

    const float S_LOG9 = 2.1972245773362196f;  // ln 9

    const int b = blockIdx.y;
    int ti = 0, rem = blockIdx.x;
    while (rem >= NTILE - ti) { rem -= NTILE - ti; ++ti; }
    const int tj  = ti + rem;
    const bool diag = (ti == tj);
    const int I = ti * TS, J = tj * TS;

    const size_t SL = (size_t)BATCH * LDIM * LDIM;       // one output slice
    const size_t mb = (size_t)b * LDIM * LDIM;
    const bool isInit = (t < 0);
    const float* ub   = u + mb;
    const float* stIn = isInit ? nullptr : (out + (size_t)t * SL + mb);
    float* vOut       = isInit ? nullptr : (out + (size_t)t * SL + mb);
    float* stOut      = (t < NSTEP - 1) ? (out + (size_t)(t + 1) * SL + mb) : nullptr;

    const int tid = threadIdx.x;
    const int j   = tid & 31;     // column within tile, varies across lanes (coalesced)
    const int i0  = tid >> 5;     // 0..7 (wave id); rows i0 + 8q

    // warm L2/L0 for the tile-1 rows used in the element loop (global_prefetch_b8)
    __builtin_prefetch(ub + (size_t)(I + i0) * LDIM + (J + j), 0, 1);
    if (!isInit)
        __builtin_prefetch(stIn + (size_t)(I + i0) * LDIM + (J + j), 0, 1);

    // ---- load tile (J,I) coalesced into LDS; transform u -> u2 on the fly ----
    #pragma unroll
    for (int q = 0; q < 4; ++q) {
        const int rr2 = i0 + 8 * q;
        const size_t g2 = (size_t)(J + rr2) * LDIM + (I + j);
        const float uv = ub[g2];
        ldsU2[rr2][j] = sigf(uv - S_LOG9) * uv;          // u2 value
        ldsA2[rr2][j] = isInit ? 0.0f : stIn[g2];
    }
    __syncthreads();

    const float4 xj = x4[(size_t)b * LDIM + (J + j)];
    const float  rj = isInit ? 0.0f : rvec[b * LDIM + (J + j)];

    float vkeep[4], nkeep[4];
    #pragma unroll
    for (int q = 0; q < 4; ++q) {
        const int i  = i0 + 8 * q;
        const int gi = I + i, gj = J + j;
        const size_t g1 = (size_t)gi * LDIM + gj;

        const float uv1 = ub[g1];
        const float u21 = sigf(uv1 - S_LOG9) * uv1;      // u2[i,j]
        const float u22 = ldsU2[j][i];                   // u2[j,i]

        const float4 xi = x4[(size_t)b * LDIM + gi];
        // m = a_i u_j + a_j u_i + c_i g_j + c_j g_i + u_i g_j + u_j g_i, banded
        float m = xi.x * xj.y + xj.x * xi.y + xi.z * xj.w + xj.z * xi.w
                + xi.y * xj.w + xj.y * xi.w;
        int d = gi - gj; d = d < 0 ? -d : d;
        if (d <= 3) m = 0.0f;

        float a1n, a2n;
        if (isInit) {
            a1n = sigf(u21) * sigf(u21 - S_LOG9);        // a_hat0[i,j]
            a2n = sigf(u22) * sigf(u22 - S_LOG9);        // a_hat0[j,i]
        } else {
            const float W  = 0.5f * (u21 + u22);         // symmetric
            const float ri = rvec[b * LDIM + gi];
            const float coef = alpha_t * m * (ri + rj - W);  // symmetric in (i,j)
            const float a1 = stIn[g1];
            const float a2 = ldsA2[j][i];
            a1n = a1 - a1 * coef;
            a2n = a2 - a2 * coef;
            a1n = fminf(fmaxf(fabsf(a1n) - alpha_next, 0.0f), 1.0f);
            a2n = fminf(fmaxf(fabsf(a2n) - alpha_next, 0.0f), 1.0f);
        }
        const float v = 0.5f * (a1n * a1n + a2n * a2n) * m;  // a_new (symmetric)
        vkeep[q] = v;
        nkeep[q] = a1n;
        ldsV[i][j] = v;
        ldsN[j][i] = a2n;    // stride-33 write: conflict-free
    }
    __syncthreads();

    // ---- stream a_new to output (non-temporal: 335 MB total, HBM-bound) ----
    if (!isInit) {
        #pragma unroll
        for (int q = 0; q < 4; ++q) {
            const int i = i0 + 8 * q;
            __builtin_nontemporal_store(vkeep[q], vOut + (size_t)(I + i) * LDIM + (J + j));
        }
        if (!diag) {
            #pragma unroll
            for (int q = 0; q < 4; ++q) {
                const int rr2 = i0 + 8 * q;
                __builtin_nontemporal_store(ldsV[j][rr2],
                                            vOut + (size_t)(J + rr2) * LDIM + (I + j));
            }
        }
    }
    // ---- next-step ah state lives in out slice t+1 (L2-resident scratch) ----
    if (stOut) {
        #pragma unroll
        for (int q = 0; q < 4; ++q) {
            const int i = i0 + 8 * q;
            stOut[(size_t)(I + i) * LDIM + (J + j)] = nkeep[q];
        }
        if (!diag) {
            #pragma unroll
            for (int q = 0; q < 4; ++q) {
                const int rr2 = i0 + 8 * q;
                stOut[(size_t)(J + rr2) * LDIM + (I + j)] = ldsN[rr2][j];
            }
        }
    }

    // ---- row/col sums of ldsV via V_WMMA_F32_16X16X4_F32 against ones ----
    // waves 0,1: row sums (rows of tile (I,J)); waves 2,3: col sums (= rows of (J,I)).
    // Branch-free A-operand addressing: no EXEC toggling around the WMMA chain.
    const int wave = tid >> 5, lane = tid & 31;
    const bool doRow = (wave < 2);
    const bool doCol = (wave >= 2 && wave < 4 && !diag);
    if (doRow || doCol) {                       // wave-uniform: EXEC all-ones inside
        v8f acc = {};
        const v2f bone = {1.0f, 1.0f};          // B = ones(4x16): layout-invariant
        const int m16 = lane & 15;
        const int kh  = (lane >> 4) << 1;       // lanes 0-15: K{0,1}; lanes 16-31: K{2,3}
        const int b16 = (wave & 1) ? 16 : 0;
        const float* lv = &ldsV[0][0];
        // A[M][K]: row sums read V[b16+m16][cb+*]; col sums read V[cb+*][b16+m16]
        const int base    = doRow ? (b16 + m16) * (TS + 1) : (b16 + m16);
        const int kstride = doRow ? 1 : (TS + 1);
        #pragma unroll
        for (int k = 0; k < 8; ++k) {
            const int a0 = base + (4 * k + kh) * kstride;
            v2f av;
            av[0] = lv[a0];
            av[1] = lv[a0 + kstride];
            acc = __builtin_amdgcn_wmma_f32_16x16x4_f32(false, av, false, bone,
                                                        (short)0, acc, false, false);
        }
        // D[M][N] = rowsum(M) replicated over N; lane0 holds M=0..7, lane16 M=8..15
        if (lane == 0 || lane == 16) {
            const int h8  = (lane == 16) ? 8 : 0;
            const int gr0 = (doRow ? I : J) + b16 + h8;
            const int oc  = doRow ? tj : ti;    // unique (row, tile) slot: deterministic
            #pragma unroll
            for (int c = 0; c < 8; ++c)
                P[((size_t)b * LDIM + (gr0 + c)) * NTILE + oc] = acc[c];
        }
    }
}

// Folds partial sums P -> S, updates lambda and r = lambda * sigmoid(S - 1).
__global__ __launch_bounds__(256)
void lambda_kernel(const float* __restrict__ P, float* __restrict__ lmbd,
                   float* __restrict__ rvec, int initPhase, float beta)
{
    const int idx = blockIdx.x * 256 + threadIdx.x;      // 0 .. 8191 = b*1024 + row
    const float* row = P + (size_t)idx * NTILE;
    float S = 0.0f;
    #pragma unroll
    for (int c = 0; c < NTILE; ++c) S += row[c];         // fixed order: deterministic
    const float z  = S - 1.0f;
    const float rl = fmaxf(z, 0.0f);
    const float lm = initPhase ? rl : (lmbd[idx] + beta * rl);
    lmbd[idx] = lm;
    rvec[idx] = lm * sigf(z);
}

extern "C" void kernel_launch(void* const* d_in, const int* in_sizes, int n_in,
                              void* d_out, int out_size, void* d_ws, size_t ws_size,
                              hipStream_t stream)
{
    (void)in_sizes; (void)n_in; (void)out_size; (void)ws_size;
    const float*  u  = (const float*)d_in[0];
    const float4* x4 = (const float4*)d_in[1];
    float* out  = (float*)d_out;

    float* P    = (float*)d_ws;                           // 8*1024*32 floats (1 MB)
    float* lmbd = P + (size_t)BATCH * LDIM * NTILE;       // 8192 floats
    float* rv   = lmbd + BATCH * LDIM;                    // 8192 floats

    dim3 grid(NPAIR, BATCH);
    const int lgrid = (BATCH * LDIM) / 256;

    // init: ah0 -> out slice 0 (scratch), row sums of a0 -> P
    fused_step_kernel<<<grid, 256, 0, stream>>>(u, x4, out, rv, P, -1, 0.0f, 0.0f);
    lambda_kernel<<<lgrid, 256, 0, stream>>>(P, lmbd, rv, 1, 0.0f);

    float alpha = 0.01f, beta = 0.1f;
    for (int t = 0; t < NSTEP; ++t) {
        fused_step_kernel<<<grid, 256, 0, stream>>>(u, x4, out, rv, P, t,
                                                    alpha, alpha * 0.99f);
        if (t < NSTEP - 1)
            lambda_kernel<<<lgrid, 256, 0, stream>>>(P, lmbd, rv, 0, beta);
        alpha *= 0.99f;
        beta  *= 0.99f;
    }
}